// HyperbolicMLR_34213709480136
// MI455X (gfx1250) — compile-verified
//
#include <hip/hip_runtime.h>
#include <hip/hip_bf16.h>
#include <math.h>

// Hyperbolic MLR (Poincare ball, c=1): reduces to two shared-B f32 GEMMs
//   PX[k,n] = <p_poincare_k, x_n>,  AX[k,n] = <a_poincare_k, x_n>
// computed with V_WMMA_F32_16X16X4_F32, async global->LDS double-buffered
// staging (ASYNCcnt), and a fused elementwise epilogue.

#define CCURV 1.0f
#define MEPS  1e-5f

typedef __attribute__((ext_vector_type(2))) float v2f;
typedef __attribute__((ext_vector_type(8))) float v8f;
typedef int v4i_ __attribute__((vector_size(16)));   // b128 payload type

// ---- CDNA5 async global->LDS path (guarded; falls back to sync staging) ----
#if defined(__has_builtin)
#if __has_builtin(__builtin_amdgcn_global_load_async_to_lds_b128)
#define USE_ASYNC 1
#endif
#endif
#ifndef USE_ASYNC
#define USE_ASYNC 0
#endif

__device__ __forceinline__ void cp_async16(const float* g, float* l) {
#if USE_ASYNC
  __builtin_amdgcn_global_load_async_to_lds_b128(
      (__attribute__((address_space(1))) v4i_*)(g),
      (__attribute__((address_space(3))) v4i_*)(l), 0, 0);
#else
  (void)g; (void)l;
#endif
}

__device__ __forceinline__ void wait_async_all() {
#if USE_ASYNC
#if __has_builtin(__builtin_amdgcn_s_wait_asynccnt)
  __builtin_amdgcn_s_wait_asynccnt(0);
#else
  asm volatile("s_wait_asynccnt 0x0" ::: "memory");
#endif
#endif
}

__device__ __forceinline__ float waveReduceSum(float v) {
#pragma unroll
  for (int off = 16; off > 0; off >>= 1)
    v += __shfl_down(v, off, 32);
  return v;
}

// ---------------- prep: expmap0(p), conformal*a, per-row scalars ----------------
__global__ void __launch_bounds__(256)
prep_rows_kernel(const float* __restrict__ a_vals,
                 const float* __restrict__ p_vals,
                 float* __restrict__ p_p, float* __restrict__ a_p,
                 float* __restrict__ g_p2, float* __restrict__ g_pa,
                 float* __restrict__ g_an, float* __restrict__ g_kk,
                 int D) {
  const int k = blockIdx.x;
  const float* u = p_vals + (size_t)k * D;
  const float* a = a_vals + (size_t)k * D;
  float su2 = 0.f, sua = 0.f, sa2 = 0.f;
  for (int d = threadIdx.x; d < D; d += blockDim.x) {
    float uu = u[d], aa = a[d];
    su2 = fmaf(uu, uu, su2);
    sua = fmaf(uu, aa, sua);
    sa2 = fmaf(aa, aa, sa2);
  }
  __shared__ float sm0[8], sm1[8], sm2[8], fin[2];
  const int lane = threadIdx.x & 31, wid = threadIdx.x >> 5;
  su2 = waveReduceSum(su2);
  sua = waveReduceSum(sua);
  sa2 = waveReduceSum(sa2);
  if (lane == 0) { sm0[wid] = su2; sm1[wid] = sua; sm2[wid] = sa2; }
  __syncthreads();
  if (threadIdx.x == 0) {
    float t0 = 0.f, t1 = 0.f, t2 = 0.f;
    for (int i = 0; i < 8; ++i) { t0 += sm0[i]; t1 += sm1[i]; t2 += sm2[i]; }
    const float sc = sqrtf(CCURV);
    float un    = fmaxf(sqrtf(t0), 1e-5f);                 // ||u||
    float th    = tanhf(fminf(fmaxf(sc * un, -15.f), 15.f));
    float scale = th / (sc * un);                          // expmap0 scale
    float p2    = scale * scale * t0;                      // ||p_poincare||^2
    float conf  = 1.f - CCURV * p2;                        // conformal factor
    float pav   = conf * scale * t1;                       // <p_p, a_p>
    float a2    = conf * conf * t2;                        // ||a_p||^2
    float an    = fmaxf(sqrtf(a2), 1e-7f);
    float lam   = 2.f / (1.f - CCURV * fminf(p2, 1.f / CCURV - 1e-4f));
    float kkv   = lam * an / sc;
    g_p2[k] = p2; g_pa[k] = pav; g_an[k] = an; g_kk[k] = kkv;
    fin[0] = scale; fin[1] = conf;
  }
  __syncthreads();
  const float scale = fin[0], conf = fin[1];
  for (int d = threadIdx.x; d < D; d += blockDim.x) {
    p_p[(size_t)k * D + d] = scale * u[d];
    a_p[(size_t)k * D + d] = conf * a[d];
  }
}

__global__ void __launch_bounds__(256)
prep_x2_kernel(const float* __restrict__ x, float* __restrict__ g_x2, int D) {
  const int n = blockIdx.x;
  const float* xr = x + (size_t)n * D;
  float s = 0.f;
  for (int d = threadIdx.x; d < D; d += blockDim.x) {
    float v = xr[d];
    s = fmaf(v, v, s);
  }
  __shared__ float sm[8];
  const int lane = threadIdx.x & 31, wid = threadIdx.x >> 5;
  s = waveReduceSum(s);
  if (lane == 0) sm[wid] = s;
  __syncthreads();
  if (threadIdx.x == 0) {
    float t = 0.f;
    for (int i = 0; i < 8; ++i) t += sm[i];
    g_x2[n] = t;
  }
}

// ---------------- main: dual-output WMMA GEMM + fused epilogue ----------------
#define BKT 32    // k rows per block
#define BNT 128   // n cols per block
#define DC  32    // depth chunk
#define LSTR 36   // LDS row stride (floats): 144B rows -> 16B-aligned B128 dsts,
                  // 36*r mod 64 distinct for r=0..15 -> bank-spread fragment reads

__device__ __forceinline__ void stage_tiles(int tid, int k0, int n0, int d0, int D,
                                            const float* __restrict__ x,
                                            const float* __restrict__ p_p,
                                            const float* __restrict__ a_p,
                                            float* sPb, float* sAb, float* sXb) {
#if USE_ASYNC
  {
    const int r = tid >> 3, c = tid & 7;      // 32 rows x 8 float4
    cp_async16(&p_p[(size_t)(k0 + r) * D + d0 + c * 4], &sPb[r * LSTR + c * 4]);
    cp_async16(&a_p[(size_t)(k0 + r) * D + d0 + c * 4], &sAb[r * LSTR + c * 4]);
  }
#pragma unroll
  for (int i = 0; i < 4; ++i) {               // 128 rows x 8 float4
    const int idx = tid + i * 256;
    const int r = idx >> 3, c = idx & 7;
    cp_async16(&x[(size_t)(n0 + r) * D + d0 + c * 4], &sXb[r * LSTR + c * 4]);
  }
#else
  {
    const int r = tid >> 3, c = tid & 7;
    const float4 vp = *(const float4*)&p_p[(size_t)(k0 + r) * D + d0 + c * 4];
    const float4 va = *(const float4*)&a_p[(size_t)(k0 + r) * D + d0 + c * 4];
    float* dp = &sPb[r * LSTR + c * 4];
    float* da = &sAb[r * LSTR + c * 4];
    dp[0] = vp.x; dp[1] = vp.y; dp[2] = vp.z; dp[3] = vp.w;
    da[0] = va.x; da[1] = va.y; da[2] = va.z; da[3] = va.w;
  }
#pragma unroll
  for (int i = 0; i < 4; ++i) {
    const int idx = tid + i * 256;
    const int r = idx >> 3, c = idx & 7;
    const float4 vx = *(const float4*)&x[(size_t)(n0 + r) * D + d0 + c * 4];
    float* dx = &sXb[r * LSTR + c * 4];
    dx[0] = vx.x; dx[1] = vx.y; dx[2] = vx.z; dx[3] = vx.w;
  }
#endif
}

__global__ void __launch_bounds__(256)
hyper_mlr_wmma_kernel(const float* __restrict__ x,
                      const float* __restrict__ p_p,
                      const float* __restrict__ a_p,
                      const float* __restrict__ g_p2,
                      const float* __restrict__ g_pa,
                      const float* __restrict__ g_an,
                      const float* __restrict__ g_kk,
                      const float* __restrict__ g_x2,
                      float* __restrict__ out,
                      int N, int K, int D) {
  __shared__ __align__(16) float sP[2][BKT * LSTR];
  __shared__ __align__(16) float sA[2][BKT * LSTR];
  __shared__ __align__(16) float sX[2][BNT * LSTR];

  const int tid  = threadIdx.x;
  const int lane = tid & 31;
  const int lo   = lane & 15;   // row within 16 (A) / col within 16 (B,C)
  const int hi   = lane >> 4;   // K-pair select (A,B) / M+8 select (C)
  const int wid  = tid >> 5;
  const int wk   = wid & 1;     // 16-row k sub-tile
  const int wn   = wid >> 1;    // 32-col n quarter

  const int n0 = blockIdx.x * BNT;
  const int k0 = blockIdx.y * BKT;

  v8f accP[2], accA[2];
#pragma unroll
  for (int j = 0; j < 2; ++j) { accP[j] = v8f{}; accA[j] = v8f{}; }

  stage_tiles(tid, k0, n0, 0, D, x, p_p, a_p, sP[0], sA[0], sX[0]);

  const int nch = D / DC;   // 16 chunks
  for (int t = 0; t < nch; ++t) {
    wait_async_all();       // my stage-t copies landed
    __syncthreads();        // everyone's landed; compute(t-1) fully retired
    if (t + 1 < nch)        // prefetch next chunk into the other buffer
      stage_tiles(tid, k0, n0, (t + 1) * DC, D, x, p_p, a_p,
                  sP[(t + 1) & 1], sA[(t + 1) & 1], sX[(t + 1) & 1]);

    const float* bP = sP[t & 1];
    const float* bA = sA[t & 1];
    const float* bX = sX[t & 1];
#pragma unroll
    for (int ds = 0; ds < 8; ++ds) {
      const int cc = ds * 4 + 2 * hi;   // fragment K-pair: {cc, cc+1}
      const v2f fp = *(const v2f*)&bP[(wk * 16 + lo) * LSTR + cc];
      const v2f fa = *(const v2f*)&bA[(wk * 16 + lo) * LSTR + cc];
#pragma unroll
      for (int j = 0; j < 2; ++j) {
        const v2f fb = *(const v2f*)&bX[(wn * 32 + j * 16 + lo) * LSTR + cc];
        accP[j] = __builtin_amdgcn_wmma_f32_16x16x4_f32(
            false, fp, false, fb, (short)0, accP[j], false, false);
        accA[j] = __builtin_amdgcn_wmma_f32_16x16x4_f32(
            false, fa, false, fb, (short)0, accA[j], false, false);
      }
    }
  }

  // ---------------- fused epilogue ----------------
  const float sc = sqrtf(CCURV);
#pragma unroll
  for (int j = 0; j < 2; ++j) {
    const int n = n0 + wn * 32 + j * 16 + lo;   // C layout: N = lane%16
    const float x2n = g_x2[n];
#pragma unroll
    for (int e = 0; e < 8; ++e) {
      const int m  = e + 8 * hi;                // C layout: M = e (+8 for hi lanes)
      const int kr = k0 + wk * 16 + m;
      const float PX = accP[j][e];
      const float AX = accA[j][e];
      const float p2k = g_p2[kr];
      const float pak = g_pa[kr];
      const float ank = g_an[kr];
      const float kv  = g_kk[kr];

      const float xy = -PX;                                     // <-p, x>
      const float al = 1.f + 2.f * CCURV * xy + CCURV * x2n;
      const float be = 1.f - CCURV * p2k;
      const float dm = 1.f + 2.f * CCURV * xy + CCURV * CCURV * p2k * x2n + MEPS;
      const float inv_dm = 1.f / dm;
      const float numdot = (al * (-pak) + be * AX) * inv_dm;     // <mob, a>
      const float num = 2.f * sc * numdot;
      const float mm = (al * al * p2k - 2.f * al * be * PX + be * be * x2n)
                       * inv_dm * inv_dm;                        // ||mob||^2
      const float den = ank * (1.f - CCURV * mm);
      const float r = num / den;
      const float ars = logf(fmaxf(r + sqrtf(fmaf(r, r, 1.f)), 1e-5f));
      out[(size_t)n * K + kr] = kv * ars;
    }
  }
}

extern "C" void kernel_launch(void* const* d_in, const int* in_sizes, int n_in,
                              void* d_out, int out_size, void* d_ws, size_t ws_size,
                              hipStream_t stream) {
  (void)n_in; (void)out_size; (void)ws_size;
  const float* x      = (const float*)d_in[0];
  const float* a_vals = (const float*)d_in[1];
  const float* p_vals = (const float*)d_in[2];
  float* out = (float*)d_out;

  const int D = 512;
  const int K = in_sizes[1] / D;   // 256
  const int N = in_sizes[0] / D;   // 2048

  float* wsf  = (float*)d_ws;
  float* p_p  = wsf;                         // K*D
  float* a_p  = p_p + (size_t)K * D;         // K*D
  float* g_p2 = a_p + (size_t)K * D;         // K
  float* g_pa = g_p2 + K;                    // K
  float* g_an = g_pa + K;                    // K
  float* g_kk = g_an + K;                    // K
  float* g_x2 = g_kk + K;                    // N

  prep_rows_kernel<<<K, 256, 0, stream>>>(a_vals, p_vals, p_p, a_p,
                                          g_p2, g_pa, g_an, g_kk, D);
  prep_x2_kernel<<<N, 256, 0, stream>>>(x, g_x2, D);

  dim3 grid(N / BNT, K / BKT);
  hyper_mlr_wmma_kernel<<<grid, 256, 0, stream>>>(x, p_p, a_p, g_p2, g_pa,
                                                  g_an, g_kk, g_x2, out, N, K, D);
}